// CtcBoundaryLossV3_77704548319397
// MI455X (gfx1250) — compile-verified
//
#include <hip/hip_runtime.h>
#include <hip/hip_bf16.h>
#include <stdint.h>

typedef float v2f __attribute__((ext_vector_type(2)));
typedef float v8f __attribute__((ext_vector_type(8)));

#define TT 2048                  // time steps (fixed by reference)
#define NTHREADS 1024            // TT/2 elements -> 2 per thread
#define MAXSPK 257               // need spikes p_0..p_256 for j<=255
#define LOG_THRESH (-1.2039728043259361f)   // ln(1 - 0.7)

// One workgroup per batch row. Computes partial[b] = sum_j |seg_j - 1| (valid j).
__global__ __launch_bounds__(NTHREADS)
void ctc_boundary_row_kernel(const float* __restrict__ alpha,
                             const float* __restrict__ ctc,   // [B,T,V], we read col 0
                             const unsigned char* __restrict__ mask,
                             const int* __restrict__ text_length,
                             float* __restrict__ partial,
                             int V)
{
    __shared__ float s_csum[TT];       // inclusive prefix sum of alpha
    __shared__ float s_fw[32];         // per-wave float totals -> exclusive offsets
    __shared__ int   s_cw[32];         // per-wave spike counts -> exclusive offsets
    __shared__ int   s_pos[MAXSPK];    // first 257 spike positions
    __shared__ float s_arr[256];       // per-interval |seg-1| (0 if invalid)
    __shared__ int   s_nspk;

    const int tid  = threadIdx.x;
    const int lane = tid & 31;
    const int wave = tid >> 5;
    const int b    = blockIdx.x;

    const int e0 = 2 * tid;
    const int e1 = e0 + 1;
    const long rowbase = (long)b * TT;

    // ---- load 2 elements per thread ----
    float a0 = alpha[rowbase + e0];
    float a1 = alpha[rowbase + e1];
    float bl0 = ctc[(rowbase + e0) * (long)V];     // strided BLANK column
    float bl1 = ctc[(rowbase + e1) * (long)V];
    int sp0 = (bl0 < LOG_THRESH) && (mask[rowbase + e0] != 0);
    int sp1 = (bl1 < LOG_THRESH) && (mask[rowbase + e1] != 0);

    // ---- wave32 inclusive scan of (pair alpha sum, pair spike count) ----
    const float fpair = a0 + a1;
    const int   cpair = sp0 + sp1;
    float fscan = fpair;
    int   cscan = cpair;
    #pragma unroll
    for (int d = 1; d < 32; d <<= 1) {
        float f = __shfl_up(fscan, d, 32);
        int   c = __shfl_up(cscan, d, 32);
        if (lane >= d) { fscan += f; cscan += c; }
    }
    if (lane == 31) { s_fw[wave] = fscan; s_cw[wave] = cscan; }
    __syncthreads();

    // ---- wave 0 scans the 32 wave totals (exclusive) ----
    if (wave == 0) {
        float fv = s_fw[lane];
        int   cv = s_cw[lane];
        float fs = fv;
        int   cs = cv;
        #pragma unroll
        for (int d = 1; d < 32; d <<= 1) {
            float f = __shfl_up(fs, d, 32);
            int   c = __shfl_up(cs, d, 32);
            if (lane >= d) { fs += f; cs += c; }
        }
        s_fw[lane] = fs - fv;   // exclusive wave offsets
        s_cw[lane] = cs - cv;
        if (lane == 31) s_nspk = cs;   // total spikes in row
    }
    __syncthreads();

    // ---- inclusive alpha cumsum + ordered spike-position scatter ----
    const float fexcl = s_fw[wave] + (fscan - fpair);  // sum alpha[0..e0-1]
    const int   cexcl = s_cw[wave] + (cscan - cpair);  // spikes before e0
    s_csum[e0] = fexcl + a0;
    s_csum[e1] = fexcl + a0 + a1;
    if (sp0 && cexcl < MAXSPK)          s_pos[cexcl]       = e0;
    if (sp1 && (cexcl + sp0) < MAXSPK)  s_pos[cexcl + sp0] = e1;
    __syncthreads();

    // ---- per-interval contributions: j in [0, 256) ----
    const int nspk = s_nspk;
    const int tlen = text_length[b];           // in [1, 256]
    if (tid < 256) {
        float v = 0.0f;
        if (tid < nspk - 1 && tid < tlen) {
            int st = s_pos[tid];
            int en = s_pos[tid + 1];
            float seg = s_csum[en] - (st > 0 ? s_csum[st - 1] : 0.0f);
            v = fabsf(seg - 1.0f);
        }
        s_arr[tid] = v;
    }
    __syncthreads();

    // ---- deterministic f32 reduction of s_arr[0..255] on the matrix pipe ----
    // 4 chained V_WMMA_F32_16X16X4_F32 with B = ones: D[m][n] = sum of all A
    // elements per row m; total = sum_m D[m][0]. Sum-of-all-A is invariant to
    // the lane->(m,k) slot permutation, so this is robust and exact (f32 FMA).
    if (wave == 0) {
        const int m  = lane & 15;
        const int kh = (lane >> 4) << 1;       // 0 or 2
        v2f bones; bones[0] = 1.0f; bones[1] = 1.0f;
        v8f acc = {};
        #pragma unroll
        for (int i = 0; i < 4; ++i) {
            v2f aA;
            aA[0] = s_arr[64 * i + 4 * m + kh];
            aA[1] = s_arr[64 * i + 4 * m + kh + 1];
            acc = __builtin_amdgcn_wmma_f32_16x16x4_f32(
                false, aA, false, bones, (short)0, acc, false, false);
        }
        float s = acc[0] + acc[1] + acc[2] + acc[3] +
                  acc[4] + acc[5] + acc[6] + acc[7];   // rows M(v) at column n(lane)
        float total = __shfl(s, 0, 32) + __shfl(s, 16, 32); // rows 0..7 + 8..15, col 0
        if (lane == 0) partial[b] = total;
    }
}

// Fixed-order final sum -> scalar loss (deterministic).
__global__ void ctc_boundary_finalize_kernel(const float* __restrict__ partial,
                                             float* __restrict__ out, int B)
{
    if (threadIdx.x == 0 && blockIdx.x == 0) {
        float s = 0.0f;
        for (int i = 0; i < B; ++i) s += partial[i];
        out[0] = s / (float)B;
    }
}

extern "C" void kernel_launch(void* const* d_in, const int* in_sizes, int n_in,
                              void* d_out, int out_size, void* d_ws, size_t ws_size,
                              hipStream_t stream) {
    const float*         alpha       = (const float*)d_in[0];
    const float*         ctc         = (const float*)d_in[1];          // [B,T,V]
    const unsigned char* mask        = (const unsigned char*)d_in[2];  // bool
    const int*           text_length = (const int*)d_in[3];
    float* out = (float*)d_out;
    float* ws  = (float*)d_ws;   // B partial sums

    const int B = in_sizes[3];                       // 16
    const int V = in_sizes[1] / in_sizes[0];         // 1024
    // T is fixed at 2048 by the reference (LDS statically sized to it).

    ctc_boundary_row_kernel<<<B, NTHREADS, 0, stream>>>(alpha, ctc, mask,
                                                        text_length, ws, V);
    ctc_boundary_finalize_kernel<<<1, 32, 0, stream>>>(ws, out, B);
}